// Momentum_69303592288702
// MI455X (gfx1250) — compile-verified
//
#include <hip/hip_runtime.h>
#include <hip/hip_bf16.h>

// ---------------------------------------------------------------------------
// Fused depthwise-conv(3x3, 12ch) + channel-reduce + Rq/Dq add, f32, gfx1250.
//
//   out[b,p,y,x] = Rq_p[b,y,x] + Dq_p[b,y,x]
//                + sum_ch sum_tap filt[ch,tap] * cache_p[b,ch,y+dy,x+dx]
//
// Decomposition: P[tap,pix] = sum_ch filt[ch,tap]*cache[ch,pix]  (a 9x12 @
// 12xN GEMM -> V_WMMA_F32_16X16X4_F32, M=taps, K=4 ch/WMMA, N=16 pixels),
// then out = 9-way shifted sum of P read from LDS. Cache tiles staged via
// async global->LDS b128 copies (ASYNCcnt path), waited with s_wait_asynccnt.
// ---------------------------------------------------------------------------

typedef float v2f __attribute__((ext_vector_type(2)));
typedef float v8f __attribute__((ext_vector_type(8)));
typedef int   v4i __attribute__((ext_vector_type(4)));

#define NXD   512
#define NYD   512
#define NCH   12
#define TW    32          // output tile width  (x)
#define THH   8           // output tile height (y)
#define TROWS 10          // rows staged: y0-1 .. y0+8
#define TCOLS 48          // cols staged: x0-8 .. x0+39  (halo [x0-1, x0+32])
#define CSTR  52          // padded col stride (bank-conflict avoidance)
#define XTILES 3          // 3 x 16-pixel WMMA column tiles cover 48 cols

#if defined(__has_builtin)
#  if __has_builtin(__builtin_amdgcn_global_load_async_to_lds_b128)
#    define HAVE_ASYNC_LDS 1
#  endif
#  if __has_builtin(__builtin_amdgcn_s_wait_asynccnt)
#    define HAVE_WAIT_ASYNC 1
#  endif
#endif

__device__ __forceinline__ void async_copy_b128(const float* g, float* l) {
#if defined(HAVE_ASYNC_LDS)
  // ROCm 7.2 prototype (from diagnostics): arg0 = v4i addrspace(1)* (global
  // src), arg1 = v4i addrspace(3)* (LDS dst), then imm offset, imm cpol.
  float* gnc = const_cast<float*>(g);
  __attribute__((address_space(1))) v4i* gp =
      (__attribute__((address_space(1))) v4i*)gnc;
  __attribute__((address_space(3))) v4i* lp =
      (__attribute__((address_space(3))) v4i*)l;
  __builtin_amdgcn_global_load_async_to_lds_b128(gp, lp, 0, 0);
#else
  float4 v = *(const float4*)g;
  *(float4*)l = v;
#endif
}

__device__ __forceinline__ void wait_async_lds() {
#if defined(HAVE_WAIT_ASYNC)
  __builtin_amdgcn_s_wait_asynccnt(0);
#elif defined(HAVE_ASYNC_LDS)
  asm volatile("s_wait_asynccnt 0" ::: "memory");
#endif
}

__global__ __launch_bounds__(256)
void momentum_grad_fused(const float* __restrict__ Rq_re, const float* __restrict__ Rq_im,
                         const float* __restrict__ Dq_re, const float* __restrict__ Dq_im,
                         const float* __restrict__ cache_re, const float* __restrict__ cache_im,
                         const float* __restrict__ filt, float* __restrict__ out)
{
  __shared__ float cache_s[NCH][TROWS][CSTR];   // staged input tile (+halo)
  __shared__ float P_s[9][TROWS][CSTR];         // per-tap partial sums

  const int tid  = threadIdx.x;
  const int lane = tid & 31;
  const int wid  = tid >> 5;
  const int x0 = blockIdx.x * TW;
  const int y0 = blockIdx.y * THH;
  const int b  = blockIdx.z;
  const bool edge = (blockIdx.x == 0) | (blockIdx.x == gridDim.x - 1) |
                    (blockIdx.y == 0) | (blockIdx.y == gridDim.y - 1);

  // ---- A operand: A[m=tap(0..8, rows 9..15 zero)][k=ch], 3 chunks of K=4.
  // 16x4 f32 A layout: lanes 0-15 hold M=lane, vgpr0=K0,vgpr1=K1;
  //                    lanes 16-31 hold M=lane-16, vgpr0=K2,vgpr1=K3.
  const int am  = lane & 15;          // tap (M row)
  const int akh = (lane >> 4) * 2;    // K half select
  v2f A[3];
#pragma unroll
  for (int j = 0; j < 3; ++j) {
    const int k0 = j * 4;
    float ax = 0.f, ay = 0.f;
    if (am < 9) {
      ax = filt[(k0 + akh + 0) * 9 + am];
      ay = filt[(k0 + akh + 1) * 9 + am];
    }
    A[j].x = ax; A[j].y = ay;
  }

#pragma unroll
  for (int p = 0; p < 2; ++p) {
    const float* cach = p ? cache_im : cache_re;
    const float* rq   = p ? Rq_im : Rq_re;
    const float* dq   = p ? Dq_im : Dq_re;

    __syncthreads();  // previous plane fully consumed LDS

    if (edge) {  // zero-fill so OOB halo taps contribute 0 (SAME padding)
      float* cz = &cache_s[0][0][0];
      for (int i = tid; i < NCH * TROWS * CSTR; i += 256) cz[i] = 0.f;
      __syncthreads();
    }

    // ---- Stage cache tile into LDS (async b128, predicated in-range) ----
    for (int g = tid; g < NCH * TROWS * (TCOLS / 4); g += 256) {
      const int ch  = g / (TROWS * (TCOLS / 4));
      const int rem = g % (TROWS * (TCOLS / 4));
      const int row = rem / (TCOLS / 4);
      const int cg  = rem % (TCOLS / 4);
      const int gy = y0 - 1 + row;
      const int gx = x0 - 8 + cg * 4;          // 16B-aligned group
      if ((unsigned)gy < NYD && (unsigned)gx < NXD) {
        const float* src = &cach[(((size_t)b * NCH + ch) * NYD + gy) * NXD + gx];
        async_copy_b128(src, &cache_s[ch][row][cg * 4]);
      }
    }
    wait_async_lds();
    __syncthreads();

    // ---- P[tap][row][col] via 3 chained V_WMMA_F32_16X16X4_F32 ----------
    // B 4x16 f32 layout: lanes 0-15 N=lane, vgpr0=K0,vgpr1=K1;
    //                    lanes 16-31 N=lane-16, vgpr0=K2,vgpr1=K3.
    const int n    = lane & 15;
    const int ksel = (lane >> 4) * 2;
    for (int task = wid; task < TROWS * XTILES; task += 8) {
      const int row     = task / XTILES;
      const int colbase = (task % XTILES) * 16;
      v8f c = {0.f, 0.f, 0.f, 0.f, 0.f, 0.f, 0.f, 0.f};
#pragma unroll
      for (int j = 0; j < 3; ++j) {
        const int kk = j * 4 + ksel;
        v2f bop;
        bop.x = cache_s[kk + 0][row][colbase + n];
        bop.y = cache_s[kk + 1][row][colbase + n];
        c = __builtin_amdgcn_wmma_f32_16x16x4_f32(
                false, A[j], false, bop, (short)0, c, false, false);
      }
      // D layout: vgpr i: lanes0-15 -> M=i, lanes16-31 -> M=i+8. Keep M<=8.
      if (lane < 16) {
#pragma unroll
        for (int i = 0; i < 8; ++i) P_s[i][row][colbase + n] = c[i];
      } else {
        P_s[8][row][colbase + n] = c[0];
      }
    }
    __syncthreads();

    // ---- 9-way shifted tap sum + Rq/Dq, store output plane --------------
    for (int idx = tid; idx < TW * THH; idx += 256) {
      const int ox = idx & (TW - 1);
      const int oy = idx / TW;
      const int gx = x0 + ox, gy = y0 + oy;
      const int col = ox + 8, row = oy + 1;
      float s = 0.f;
#pragma unroll
      for (int t = 0; t < 9; ++t) {
        const int dy = t / 3 - 1, dx = t % 3 - 1;
        s += P_s[t][row + dy][col + dx];
      }
      const size_t pix = (size_t)b * (NYD * NXD) + (size_t)gy * NXD + gx;
      s += rq[pix] + dq[pix];
      out[((size_t)b * 2 + p) * (size_t)(NYD * NXD) + (size_t)gy * NXD + gx] = s;
    }
  }
}

extern "C" void kernel_launch(void* const* d_in, const int* in_sizes, int n_in,
                              void* d_out, int out_size, void* d_ws, size_t ws_size,
                              hipStream_t stream) {
  (void)n_in; (void)out_size; (void)d_ws; (void)ws_size;
  const float* Rq_re    = (const float*)d_in[0];
  const float* Rq_im    = (const float*)d_in[1];
  const float* Dq_re    = (const float*)d_in[2];
  const float* Dq_im    = (const float*)d_in[3];
  const float* cache_re = (const float*)d_in[4];
  const float* cache_im = (const float*)d_in[5];
  const float* filt     = (const float*)d_in[6];
  float* out = (float*)d_out;

  const int B = in_sizes[0] / (NYD * NXD);   // (B,1,Ny,Nx) plane count
  dim3 grid(NXD / TW, NYD / THH, B);
  momentum_grad_fused<<<grid, 256, 0, stream>>>(
      Rq_re, Rq_im, Dq_re, Dq_im, cache_re, cache_im, filt, out);
}